// GraphConvolutionHypernetwork_31172872634853
// MI455X (gfx1250) — compile-verified
//
#include <hip/hip_runtime.h>
#include <math.h>

#define BSZ    2048
#define NNODE  128
#define EDIM   64
#define FDIM   256
#define NLAYER 2

typedef float v2f __attribute__((ext_vector_type(2)));
typedef float v8f __attribute__((ext_vector_type(8)));

// ---- workspace layout (floats) ----
// Region A: per-batch hyper vectors, stride 640 floats
#define WSA_STRIDE 640
#define WSA_W1 0      // |w1|            (64)
#define WSA_B1 64     // b1              (64)
#define WSA_BB 128    // layer bias Bb_i (64 each, i<2)
#define WSA_W2 256    // |w2|            (128)
#define WSA_B2 384    // b2              (64)
#define WSA_W3 448    // |w3|            (64)
#define WSA_B3 512    // b3 scalar
// Region B: generated layer weights Wmat[b][layer][64*64]
#define WSB_OFF ((size_t)BSZ * WSA_STRIDE)

// ============================================================
// Kernel 1: hyper-network. One block = 16 batches, 256 threads.
// Weights are streamed once per 16 batches (128 blocks total),
// so layerW_fc2_W (2 MB) costs ~256 MB of L2 traffic instead of 4 GB.
// ============================================================
__global__ __launch_bounds__(256) void hyper_kernel(
    const float* __restrict__ hx,
    const float* __restrict__ w1_W,  const float* __restrict__ w1_b,
    const float* __restrict__ b1_W,  const float* __restrict__ b1_b,
    const float* __restrict__ lW1_W, const float* __restrict__ lW1_b,
    const float* __restrict__ lW2_W, const float* __restrict__ lW2_b,
    const float* __restrict__ bW1_W, const float* __restrict__ bW1_b,
    const float* __restrict__ bW2_W, const float* __restrict__ bW2_b,
    const float* __restrict__ w2_W,  const float* __restrict__ w2_b,
    const float* __restrict__ b2_W,  const float* __restrict__ b2_b,
    const float* __restrict__ w3_W,  const float* __restrict__ w3_b,
    const float* __restrict__ b3f1_W,const float* __restrict__ b3f1_b,
    const float* __restrict__ b3f2_W,const float* __restrict__ b3f2_b,
    float* __restrict__ ws)
{
  __shared__ float s_hx[16][FDIM];
  __shared__ float s_hid[16][EDIM];
  __shared__ float s_tmp[16][EDIM];

  const int t  = threadIdx.x;
  const int b0 = blockIdx.x * 16;

  for (int i = t; i < 16 * FDIM; i += 256)
    s_hx[0][i] = hx[(size_t)b0 * FDIM + i];
  __syncthreads();

  const int u  = t & 63;   // output index within a 64-vector
  const int g4 = t >> 6;   // this thread covers batches g4, g4+4, g4+8, g4+12

  float v0, v1, v2, v3;
  auto gemvF = [&](const float* W, const float* bia) {
    float a0 = bia[u], a1 = a0, a2 = a0, a3 = a0;
    for (int f = 0; f < FDIM; ++f) {
      float w = W[f * EDIM + u];
      a0 = fmaf(s_hx[g4][f],    w, a0);
      a1 = fmaf(s_hx[g4+4][f],  w, a1);
      a2 = fmaf(s_hx[g4+8][f],  w, a2);
      a3 = fmaf(s_hx[g4+12][f], w, a3);
    }
    v0 = a0; v1 = a1; v2 = a2; v3 = a3;
  };
  auto storeA = [&](int off, bool doabs) {
    ws[((size_t)(b0+g4)   )*WSA_STRIDE + off + u] = doabs ? fabsf(v0) : v0;
    ws[((size_t)(b0+g4+4) )*WSA_STRIDE + off + u] = doabs ? fabsf(v1) : v1;
    ws[((size_t)(b0+g4+8) )*WSA_STRIDE + off + u] = doabs ? fabsf(v2) : v2;
    ws[((size_t)(b0+g4+12))*WSA_STRIDE + off + u] = doabs ? fabsf(v3) : v3;
  };

  gemvF(w1_W, w1_b); storeA(WSA_W1, true);
  gemvF(b1_W, b1_b); storeA(WSA_B1, false);
  gemvF(b2_W, b2_b); storeA(WSA_B2, false);
  gemvF(w3_W, w3_b); storeA(WSA_W3, true);

  // w2: F -> 128, two halves of 64
  for (int h = 0; h < 2; ++h) {
    float a0 = w2_b[u + 64*h], a1 = a0, a2 = a0, a3 = a0;
    for (int f = 0; f < FDIM; ++f) {
      float w = w2_W[f * NNODE + u + 64*h];
      a0 = fmaf(s_hx[g4][f],    w, a0);
      a1 = fmaf(s_hx[g4+4][f],  w, a1);
      a2 = fmaf(s_hx[g4+8][f],  w, a2);
      a3 = fmaf(s_hx[g4+12][f], w, a3);
    }
    v0 = a0; v1 = a1; v2 = a2; v3 = a3;
    storeA(WSA_W2 + 64*h, true);
  }

  // b3 = relu(h_x @ fc1 + b) @ fc2 + b
  gemvF(b3f1_W, b3f1_b);
  s_tmp[g4][u]    = fmaxf(v0, 0.f);
  s_tmp[g4+4][u]  = fmaxf(v1, 0.f);
  s_tmp[g4+8][u]  = fmaxf(v2, 0.f);
  s_tmp[g4+12][u] = fmaxf(v3, 0.f);
  __syncthreads();
  if (t < 16) {
    float a = b3f2_b[0];
    for (int e = 0; e < EDIM; ++e) a = fmaf(s_tmp[t][e], b3f2_W[e], a);
    ws[((size_t)(b0+t))*WSA_STRIDE + WSA_B3] = a;
  }
  __syncthreads();

  for (int i = 0; i < NLAYER; ++i) {
    // hidden for layer bias MLP
    gemvF(bW1_W + (size_t)i*FDIM*EDIM, bW1_b + i*EDIM);
    s_tmp[g4][u]    = fmaxf(v0, 0.f);
    s_tmp[g4+4][u]  = fmaxf(v1, 0.f);
    s_tmp[g4+8][u]  = fmaxf(v2, 0.f);
    s_tmp[g4+12][u] = fmaxf(v3, 0.f);
    // hidden for layer weight MLP
    gemvF(lW1_W + (size_t)i*FDIM*EDIM, lW1_b + i*EDIM);
    s_hid[g4][u]    = fmaxf(v0, 0.f);
    s_hid[g4+4][u]  = fmaxf(v1, 0.f);
    s_hid[g4+8][u]  = fmaxf(v2, 0.f);
    s_hid[g4+12][u] = fmaxf(v3, 0.f);
    __syncthreads();

    // Bb_i = hidB @ bW2[i] + bb2[i]
    {
      const float* W = bW2_W + (size_t)i*EDIM*EDIM;
      float a0 = bW2_b[i*EDIM + u], a1 = a0, a2 = a0, a3 = a0;
      for (int e = 0; e < EDIM; ++e) {
        float w = W[e * EDIM + u];
        a0 = fmaf(s_tmp[g4][e],    w, a0);
        a1 = fmaf(s_tmp[g4+4][e],  w, a1);
        a2 = fmaf(s_tmp[g4+8][e],  w, a2);
        a3 = fmaf(s_tmp[g4+12][e], w, a3);
      }
      v0 = a0; v1 = a1; v2 = a2; v3 = a3;
      storeA(WSA_BB + 64*i, false);
    }

    // Wmat_i = |hidW @ lW2[i] + lb2[i]|  : 16 batches per weight read
    const float* W2 = lW2_W + (size_t)i*EDIM*(EDIM*EDIM);
    const float* B2 = lW2_b + (size_t)i*(EDIM*EDIM);
    for (int r = 0; r < 16; ++r) {
      int o = t + 256*r;
      float bias = B2[o];
      float acc[16];
      #pragma unroll
      for (int g = 0; g < 16; ++g) acc[g] = bias;
      for (int e = 0; e < EDIM; ++e) {
        float w = W2[(size_t)e*(EDIM*EDIM) + o];
        #pragma unroll
        for (int g = 0; g < 16; ++g) acc[g] = fmaf(s_hid[g][e], w, acc[g]);
      }
      #pragma unroll
      for (int g = 0; g < 16; ++g)
        ws[WSB_OFF + (((size_t)(b0+g))*NLAYER + i)*(EDIM*EDIM) + o] = fabsf(acc[g]);
    }
    __syncthreads();
  }
}

// ============================================================
// Kernel 2: per-batch forward with U·Uᵀ = I collapse.
// One block = one batch, 256 threads = 8 waves.
// Layer GEMM x(128x64) @ W(64x64) via V_WMMA_F32_16X16X4_F32:
// wave w owns M-tile w (rows 16w..16w+15), iterates 4 N-tiles x 16 K-steps.
// s_x stride 68 (pad 4): wave-wide b64 loads hit banks 4*lo+2*hi+{0,1}
//   -> all 64 banks exactly once, conflict-free.
// W stored TRANSPOSED in LDS (s_Wt[col][k]) so each B fragment
//   (W[k][col], W[k+1][col]) is one contiguous ds_load_b64.
// ============================================================
__global__ __launch_bounds__(256) void gconv_main_kernel(
    const float* __restrict__ x_in,
    const float* __restrict__ ws,
    float* __restrict__ out)
{
  constexpr int XP  = EDIM + 4;  // 68: conflict-free padded stride
  constexpr int WTP = EDIM + 4;  // 68
  __shared__ float s_x[NNODE * XP];
  __shared__ float s_Wt[EDIM * WTP];   // transposed: [col][k]
  __shared__ float s_pB[EDIM];
  __shared__ float s_w2[NNODE];
  __shared__ float s_outx[EDIM];

  const int t = threadIdx.x;
  const int b = blockIdx.x;
  const float* Aw = ws + (size_t)b * WSA_STRIDE;

  // input projection: x0[n,e] = x[n] * |w1|[e] + b1[e]
  for (int idx = t; idx < NNODE * EDIM; idx += 256) {
    int n = idx >> 6, e = idx & 63;
    s_x[n * XP + e] = x_in[(size_t)b * NNODE + n] * Aw[WSA_W1 + e] + Aw[WSA_B1 + e];
  }

  const int lane = t & 31, lo = lane & 15, hi = lane >> 4;
  const int m0 = (t >> 5) * 16;

  for (int i = 0; i < NLAYER; ++i) {
    const float* Wg = ws + WSB_OFF + ((size_t)b * NLAYER + i) * (EDIM*EDIM);
    // coalesced global read, transposed LDS store: s_Wt[col][k] = W[k][col]
    for (int idx = t; idx < EDIM*EDIM; idx += 256)
      s_Wt[(idx & 63) * WTP + (idx >> 6)] = Wg[idx];
    if (t < EDIM) s_pB[t] = Aw[WSA_BB + 64*i + t];
    __syncthreads();

    // A layout (16x4 f32): lanes 0-15 = M, VGPR0/1 = K, lane-half selects K/K+2
    v8f acc[4] = {};
    const float* xrow = &s_x[(m0 + lo) * XP + 2*hi];
    const float* wcol = &s_Wt[lo * WTP + 2*hi];
    #pragma unroll
    for (int kk = 0; kk < EDIM; kk += 4) {
      v2f a = *(const v2f*)&xrow[kk];                 // x[m0+lo, kk+2hi .. +1]
      #pragma unroll
      for (int nt = 0; nt < 4; ++nt) {
        v2f bv = *(const v2f*)&wcol[nt * 16 * WTP + kk];  // W[kk+2hi..+1, nt*16+lo]
        acc[nt] = __builtin_amdgcn_wmma_f32_16x16x4_f32(
            false, a, false, bv, (short)0, acc[nt], false, false);
      }
    }

    // writeback with bias + elu; wave writes only its own rows
    #pragma unroll
    for (int nt = 0; nt < 4; ++nt) {
      int col = nt * 16 + lo;
      float bb = s_pB[col];
      #pragma unroll
      for (int j = 0; j < 8; ++j) {
        float z = acc[nt][j] + bb;
        z = z > 0.f ? z : (__expf(z) - 1.f);
        s_x[(m0 + j + 8*hi) * XP + col] = z;
      }
    }
    __syncthreads();
  }

  // out_x[e] = sum_n x[n,e]*|w2|[n] + b2[e];  y = sum_e out_x[e]*|w3|[e] + b3
  if (t < NNODE) s_w2[t] = Aw[WSA_W2 + t];
  __syncthreads();
  if (t < EDIM) {
    float a = Aw[WSA_B2 + t];
    for (int n = 0; n < NNODE; ++n)
      a = fmaf(s_x[n * XP + t], s_w2[n], a);
    s_outx[t] = a * Aw[WSA_W3 + t];
  }
  __syncthreads();
  if (t == 0) {
    float y = Aw[WSA_B3];
    for (int e = 0; e < EDIM; ++e) y += s_outx[e];
    out[b] = y;
  }
}

extern "C" void kernel_launch(void* const* d_in, const int* in_sizes, int n_in,
                              void* d_out, int out_size, void* d_ws, size_t ws_size,
                              hipStream_t stream) {
  const float* x_in   = (const float*)d_in[0];
  // d_in[1] = adj: unused. eigh(L) gives orthonormal U, and it appears only as
  // U @ ((U^T x) @ W) = (U U^T) x W = x W  (permutation reorder also cancels).
  const float* hx     = (const float*)d_in[2];
  const float* w1_W   = (const float*)d_in[3];
  const float* w1_b   = (const float*)d_in[4];
  const float* b1_W   = (const float*)d_in[5];
  const float* b1_b   = (const float*)d_in[6];
  const float* lW1_W  = (const float*)d_in[7];
  const float* lW1_b  = (const float*)d_in[8];
  const float* lW2_W  = (const float*)d_in[9];
  const float* lW2_b  = (const float*)d_in[10];
  const float* bW1_W  = (const float*)d_in[11];
  const float* bW1_b  = (const float*)d_in[12];
  const float* bW2_W  = (const float*)d_in[13];
  const float* bW2_b  = (const float*)d_in[14];
  const float* w2_W   = (const float*)d_in[15];
  const float* w2_b   = (const float*)d_in[16];
  const float* b2_W   = (const float*)d_in[17];
  const float* b2_b   = (const float*)d_in[18];
  const float* w3_W   = (const float*)d_in[19];
  const float* w3_b   = (const float*)d_in[20];
  const float* b3f1_W = (const float*)d_in[21];
  const float* b3f1_b = (const float*)d_in[22];
  const float* b3f2_W = (const float*)d_in[23];
  const float* b3f2_b = (const float*)d_in[24];
  float* ws  = (float*)d_ws;   // needs ~70 MB: 5.25 MB vectors + 64 MB Wmat
  float* out = (float*)d_out;

  hyper_kernel<<<BSZ/16, 256, 0, stream>>>(
      hx, w1_W, w1_b, b1_W, b1_b, lW1_W, lW1_b, lW2_W, lW2_b,
      bW1_W, bW1_b, bW2_W, bW2_b, w2_W, w2_b, b2_W, b2_b,
      w3_W, w3_b, b3f1_W, b3f1_b, b3f2_W, b3f2_b, ws);

  gconv_main_kernel<<<BSZ, 256, 0, stream>>>(x_in, ws, out);
}